// EMAVectorQuantizer_429496730211
// MI455X (gfx1250) — compile-verified
//
#include <hip/hip_runtime.h>

typedef __attribute__((ext_vector_type(2))) float v2f;
typedef __attribute__((ext_vector_type(8))) float v8f;

#define N_PIX     16384      // B*H*W = 16*32*32
#define C_DIM     256
#define K_EMB     8192
#define HW_DIM    1024       // H*W
#define OUT_ELEMS 4194304    // 16*256*32*32
#define KSPLIT    4          // K dimension split for occupancy
#define KT_TILES  (K_EMB / 16)          // 512 candidate tiles
#define KT_PER    (KT_TILES / KSPLIT)   // 128 tiles per chunk

// ---------------------------------------------------------------------------
// Kernel 1: e_norms[k] = sum_c E[k,c]^2.  One wave per codebook row.
// ---------------------------------------------------------------------------
__global__ __launch_bounds__(256) void vq_enorm_kernel(const float* __restrict__ emb,
                                                       float* __restrict__ enorms) {
    const int wave = threadIdx.x >> 5;
    const int lane = threadIdx.x & 31;
    const int k    = blockIdx.x * 8 + wave;
    const float4* row = (const float4*)(emb + (size_t)k * C_DIM);
    float4 v0 = row[lane];
    float4 v1 = row[lane + 32];
    float s = v0.x * v0.x + v0.y * v0.y + v0.z * v0.z + v0.w * v0.w +
              v1.x * v1.x + v1.y * v1.y + v1.z * v1.z + v1.w * v1.w;
#pragma unroll
    for (int off = 16; off > 0; off >>= 1) s += __shfl_xor(s, off, 32);
    if (lane == 0) enorms[k] = s;
}

// ---------------------------------------------------------------------------
// Kernel 2: fused fp32 WMMA GEMM (F @ E^T) + running argmin over a K chunk.
// One wave = one 16-row M tile x 2048 candidates.  A (16x256) lives in 128
// VGPRs, laid out for V_WMMA_F32_16X16X4_F32: lanes 0-15 hold K={0,1},
// lanes 16-31 hold K={2,3} of each 4-wide K step.  B streams as float2 per
// lane; next tile is prefetched (global_prefetch_b8) one iteration ahead.
// Score minimized: ||e_k||^2 - 2*f.e_k  (||f||^2 is constant per row).
// ---------------------------------------------------------------------------
__global__ __launch_bounds__(256) void vq_argmin_kernel(const float* __restrict__ in,
                                                        const float* __restrict__ emb,
                                                        const float* __restrict__ enorms,
                                                        float* __restrict__ pbestv,
                                                        int* __restrict__ pbestk) {
    const int lane   = threadIdx.x & 31;
    const int wave   = threadIdx.x >> 5;
    const int mn     = lane & 15;     // row within tile (A) / candidate col (B,C)
    const int half   = lane >> 4;     // selects K pair (A,B) / row group (C)
    const int mgroup = blockIdx.x & (N_PIX / 128 - 1);   // 128 M-groups
    const int chunk  = blockIdx.x >> 7;                  // 4 K-chunks
    const int m_base = (mgroup * 8 + wave) * 16;

    // Preload A slab: F[m, c] = inputs[b, c, hw]; coalesced across lanes 0-15.
    const int m  = m_base + mn;
    const int bb = m >> 10;
    const int hw = m & (HW_DIM - 1);
    const float* ap = in + (size_t)bb * (C_DIM * HW_DIM) + hw;

    v2f a[64];
#pragma unroll
    for (int t = 0; t < 64; ++t) {
        const int c0 = 4 * t + 2 * half;
        a[t].x = ap[(size_t)c0 * HW_DIM];
        a[t].y = ap[(size_t)(c0 + 1) * HW_DIM];
    }

    float bestv[8];
    int   bestk[8];
#pragma unroll
    for (int r = 0; r < 8; ++r) { bestv[r] = 3.4e38f; bestk[r] = 0; }

    const int kt_begin = chunk * KT_PER;
    const int kt_end   = kt_begin + KT_PER;
    for (int kt = kt_begin; kt < kt_end; ++kt) {
        // B operand: lane (half,mn) needs E[kt*16+mn, 4t + 2*half + {0,1}]
        const float* ep = emb + (size_t)(kt * 16 + mn) * C_DIM + 2 * half;
        if (kt + 1 < kt_end) {
            // pull next tile's row of B into WGP$/L2 one iteration early
            __builtin_prefetch(ep + 16 * C_DIM, 0, 0);
        }
        v8f acc = {};
#pragma unroll
        for (int t = 0; t < 64; ++t) {
            v2f bv = *(const v2f*)(ep + 4 * t);   // 8B aligned, contiguous
            acc = __builtin_amdgcn_wmma_f32_16x16x4_f32(
                false, a[t], false, bv, (short)0, acc, false, false);
        }
        const int   kcand = kt * 16 + mn;     // C/D layout: N = lane & 15
        const float en    = enorms[kcand];
#pragma unroll
        for (int r = 0; r < 8; ++r) {         // row = m_base + r + 8*half
            const float s = en - 2.0f * acc[r];
            if (s < bestv[r] || (s == bestv[r] && kcand < bestk[r])) {
                bestv[r] = s; bestk[r] = kcand;
            }
        }
    }

    // Min-reduce across the 16 candidate classes (stay within each lane-half).
#pragma unroll
    for (int off = 1; off < 16; off <<= 1) {
#pragma unroll
        for (int r = 0; r < 8; ++r) {
            float ov = __shfl_xor(bestv[r], off, 32);
            int   oi = __shfl_xor(bestk[r], off, 32);
            if (ov < bestv[r] || (ov == bestv[r] && oi < bestk[r])) {
                bestv[r] = ov; bestk[r] = oi;
            }
        }
    }
    if (mn == 0) {
#pragma unroll
        for (int r = 0; r < 8; ++r) {
            const int row = m_base + half * 8 + r;
            pbestv[chunk * N_PIX + row] = bestv[r];
            pbestk[chunk * N_PIX + row] = bestk[r];
        }
    }
}

// ---------------------------------------------------------------------------
// Kernel 3: combine the KSPLIT partial argmins (index tie-break keeps
// first-occurrence semantics of jnp.argmin).
// ---------------------------------------------------------------------------
__global__ __launch_bounds__(256) void vq_combine_kernel(const float* __restrict__ pbestv,
                                                         const int* __restrict__ pbestk,
                                                         int* __restrict__ idx) {
    const int n = blockIdx.x * 256 + threadIdx.x;
    float bv = pbestv[n];
    int   bk = pbestk[n];
#pragma unroll
    for (int c = 1; c < KSPLIT; ++c) {
        const float v = pbestv[c * N_PIX + n];
        const int   k = pbestk[c * N_PIX + n];
        if (v < bv || (v == bv && k < bk)) { bv = v; bk = k; }
    }
    idx[n] = bk;
}

// ---------------------------------------------------------------------------
// Kernel 4: gather quantized output (BCHW order, coalesced), emit idx as
// float, and produce deterministic per-block partial sums of (q - x)^2.
// ---------------------------------------------------------------------------
__global__ __launch_bounds__(256) void vq_gather_kernel(const float* __restrict__ in,
                                                        const float* __restrict__ emb,
                                                        const int* __restrict__ idx,
                                                        float* __restrict__ out,
                                                        float* __restrict__ idxf,
                                                        float* __restrict__ partials) {
    const int gid = blockIdx.x * 256 + threadIdx.x;
    const int b   = gid >> 18;          // / (C*HW)
    const int rem = gid & 262143;
    const int c   = rem >> 10;
    const int hw  = rem & 1023;
    const int n   = (b << 10) | hw;
    const int k   = idx[n];
    const float q = emb[(size_t)k * C_DIM + c];
    const float x = in[gid];
    out[gid] = q;                        // straight-through value == quantized
    if (c == 0) idxf[n] = (float)k;
    float d = q - x;
    d = d * d;

    __shared__ float s[256];
    s[threadIdx.x] = d;
    __syncthreads();
#pragma unroll
    for (int st = 128; st > 0; st >>= 1) {
        if (threadIdx.x < st) s[threadIdx.x] += s[threadIdx.x + st];
        __syncthreads();
    }
    if (threadIdx.x == 0) partials[blockIdx.x] = s[0];
}

// ---------------------------------------------------------------------------
// Kernel 5: final deterministic reduction -> loss = 0.25 * mean(diff^2)
// ---------------------------------------------------------------------------
__global__ __launch_bounds__(256) void vq_loss_kernel(const float* __restrict__ partials,
                                                      float* __restrict__ loss) {
    __shared__ float s[256];
    float t = 0.0f;
    for (int j = 0; j < 64; ++j) t += partials[threadIdx.x + 256 * j];
    s[threadIdx.x] = t;
    __syncthreads();
#pragma unroll
    for (int st = 128; st > 0; st >>= 1) {
        if (threadIdx.x < st) s[threadIdx.x] += s[threadIdx.x + st];
        __syncthreads();
    }
    if (threadIdx.x == 0) loss[0] = 0.25f * s[0] / (float)OUT_ELEMS;
}

// ---------------------------------------------------------------------------
extern "C" void kernel_launch(void* const* d_in, const int* in_sizes, int n_in,
                              void* d_out, int out_size, void* d_ws, size_t ws_size,
                              hipStream_t stream) {
    (void)in_sizes; (void)n_in; (void)out_size; (void)ws_size;
    const float* in  = (const float*)d_in[0];   // [16,256,32,32] f32
    const float* emb = (const float*)d_in[1];   // [8192,256] f32

    float* out  = (float*)d_out;                // 4194304 quantized values
    float* loss = out + OUT_ELEMS;              // 1 scalar
    float* idxf = out + OUT_ELEMS + 1;          // 16384 indices (as float)

    char*  ws       = (char*)d_ws;
    float* enorms   = (float*)ws;               // 8192  f32  ( 32 KB)
    float* pbestv   = (float*)(ws + 32768);     // 4*16384 f32 (256 KB)
    int*   pbestk   = (int*)(ws + 294912);      // 4*16384 i32 (256 KB)
    int*   idx      = (int*)(ws + 557056);      // 16384 i32  ( 64 KB)
    float* partials = (float*)(ws + 622592);    // 16384 f32  ( 64 KB)

    vq_enorm_kernel  <<<K_EMB / 8,              256, 0, stream>>>(emb, enorms);
    vq_argmin_kernel <<<(N_PIX / 128) * KSPLIT, 256, 0, stream>>>(in, emb, enorms,
                                                                  pbestv, pbestk);
    vq_combine_kernel<<<N_PIX / 256,            256, 0, stream>>>(pbestv, pbestk, idx);
    vq_gather_kernel <<<OUT_ELEMS / 256,        256, 0, stream>>>(in, emb, idx, out,
                                                                  idxf, partials);
    vq_loss_kernel   <<<1,                      256, 0, stream>>>(partials, loss);
}